// ISSDiagonalizer_32057635897897
// MI455X (gfx1250) — compile-verified
//
#include <hip/hip_runtime.h>
#include <math.h>

typedef __attribute__((ext_vector_type(2))) float v2f;
typedef __attribute__((ext_vector_type(4))) float v4f;
typedef __attribute__((ext_vector_type(8))) float v8f;

#define B_ 8
#define F_ 513
#define M_ 8
#define T_ 512
#define EPS_  1e-6f
#define EPS2_ 1e-6f
#define EPS3_ 1e-3f
#define N_ITER_ 2

#define QN_  ((size_t)B_ * F_ * M_ * M_ * 2)   // complex Q as interleaved re,im
#define XTN_ ((size_t)B_ * F_ * M_ * T_)
#define MT4_ (M_ * T_ / 4)                     // 1024 float4 per (b,f) plane

// ---------------------------------------------------------------------------
// Kernel 1: one workgroup (256 thr = 8 waves) per (b,f).
//   Phase A: stage x_re, x_im, w=max(r,eps3), Q into LDS (B128 NT loads).
//   Phase B: V = (1/T) sum_t w * x * conj(x) via f32 WMMA (two real GEMMs,
//            64x512 @ 512x16, 16x16 tile per wave, K=4 per WMMA,
//            4 independent accumulator chains).
//   Phase C: diagonal loading with trace.
//   Phase D: ISS sweeps (2 iters x 8 pivots) on wave 0 (wave32 lockstep).
//   Phase E: Qx, xt=|Qx|^2 (unnormalized, RT store -> stays in L2);
//            per-(b,f) sum -> d_ws.
// ---------------------------------------------------------------------------
__global__ __launch_bounds__(256) void iss_main_kernel(
    const float* __restrict__ g_r,
    const float* __restrict__ g_xr,
    const float* __restrict__ g_xi,
    const float* __restrict__ g_Qr,
    const float* __restrict__ g_Qi,
    float* __restrict__ outQ,     // [B*F*64*2] interleaved re,im (unnormalized)
    float* __restrict__ outXT,    // [B*F*8*512] (unnormalized)
    float* __restrict__ partial)  // [B*F] per-(b,f) sum of xt
{
    __shared__ float s_w [M_][T_];
    __shared__ float s_xr[M_][T_];
    __shared__ float s_xi[M_][T_];
    __shared__ float s_stage[8][16][16];
    __shared__ float s_Vre[M_][M_][M_];
    __shared__ float s_Vim[M_][M_][M_];
    __shared__ float s_Qr[M_][M_];
    __shared__ float s_Qi[M_][M_];
    __shared__ float s_Vqr[M_][M_];
    __shared__ float s_Vqi[M_][M_];
    __shared__ float s_vr[M_];
    __shared__ float s_vi[M_];
    __shared__ float s_red[256];

    const int bf  = blockIdx.x;          // b*F + f
    const int tid = threadIdx.x;
    const int wave = tid >> 5;
    const int lane = tid & 31;

    // ---------------- Phase A: stage inputs (vectorized, non-temporal) ------
    {
        const size_t base4 = (size_t)bf * MT4_;
        const v4f* g4r  = (const v4f*)g_r;
        const v4f* g4xr = (const v4f*)g_xr;
        const v4f* g4xi = (const v4f*)g_xi;
        v4f* fw  = (v4f*)&s_w[0][0];
        v4f* fxr = (v4f*)&s_xr[0][0];
        v4f* fxi = (v4f*)&s_xi[0][0];
        for (int i = tid; i < MT4_; i += 256) {
            v4f rv = __builtin_nontemporal_load(&g4r[base4 + i]);
            rv.x = fmaxf(rv.x, EPS3_);
            rv.y = fmaxf(rv.y, EPS3_);
            rv.z = fmaxf(rv.z, EPS3_);
            rv.w = fmaxf(rv.w, EPS3_);
            fw[i]  = rv;
            fxr[i] = __builtin_nontemporal_load(&g4xr[base4 + i]);
            fxi[i] = __builtin_nontemporal_load(&g4xi[base4 + i]);
        }
        if (tid < 64) {
            s_Qr[tid >> 3][tid & 7] =
                __builtin_nontemporal_load(&g_Qr[(size_t)bf * 64 + tid]);
            s_Qi[tid >> 3][tid & 7] =
                __builtin_nontemporal_load(&g_Qi[(size_t)bf * 64 + tid]);
        }
    }
    __syncthreads();

    // ---------------- Phase B: WMMA covariance GEMMs ----------------
    {
        const int combo = wave >> 2;      // 0: A = w*x_re, 1: A = w*x_im
        const int rt    = wave & 3;       // row tile of the 64-row (k,m) space
        const int hl    = lane >> 4;      // lane half (selects K pair / row group)
        const int l16   = lane & 15;
        const int row   = rt * 16 + l16;  // packed (k,m)
        const int kA    = row >> 3;
        const int mA    = row & 7;
        const int nB    = l16;            // B column: 0..7 -> x_re, 8..15 -> x_im

        const float* wrow = &s_w[kA][0];
        const float* arow = combo ? &s_xi[mA][0] : &s_xr[mA][0];
        const float* brow = (nB < 8) ? &s_xr[nB][0] : &s_xi[nB - 8][0];

        v8f acc0 = {}; v8f acc1 = {}; v8f acc2 = {}; v8f acc3 = {};
        for (int i = 0; i < 32; ++i) {
            const int tb = i * 4 + hl * 2;
            v2f a, b;

            a.x = wrow[tb]       * arow[tb];
            a.y = wrow[tb + 1]   * arow[tb + 1];
            b.x = brow[tb];
            b.y = brow[tb + 1];
            acc0 = __builtin_amdgcn_wmma_f32_16x16x4_f32(false, a, false, b,
                                                         (short)0, acc0, false, false);

            a.x = wrow[128 + tb]     * arow[128 + tb];
            a.y = wrow[128 + tb + 1] * arow[128 + tb + 1];
            b.x = brow[128 + tb];
            b.y = brow[128 + tb + 1];
            acc1 = __builtin_amdgcn_wmma_f32_16x16x4_f32(false, a, false, b,
                                                         (short)0, acc1, false, false);

            a.x = wrow[256 + tb]     * arow[256 + tb];
            a.y = wrow[256 + tb + 1] * arow[256 + tb + 1];
            b.x = brow[256 + tb];
            b.y = brow[256 + tb + 1];
            acc2 = __builtin_amdgcn_wmma_f32_16x16x4_f32(false, a, false, b,
                                                         (short)0, acc2, false, false);

            a.x = wrow[384 + tb]     * arow[384 + tb];
            a.y = wrow[384 + tb + 1] * arow[384 + tb + 1];
            b.x = brow[384 + tb];
            b.y = brow[384 + tb + 1];
            acc3 = __builtin_amdgcn_wmma_f32_16x16x4_f32(false, a, false, b,
                                                         (short)0, acc3, false, false);
        }
        v8f acc = acc0 + acc1 + acc2 + acc3;
        // D layout: VGPR j -> M=j (lanes 0-15) / M=j+8 (lanes 16-31), N = lane%16
#pragma unroll
        for (int j = 0; j < 8; ++j)
            s_stage[wave][hl * 8 + j][l16] = acc[j];
    }
    __syncthreads();

    // ---------------- Assemble complex V, scale by 1/T ----------------
    for (int p = tid; p < M_ * M_ * M_; p += 256) {
        const int k = p >> 6;
        const int m = (p >> 3) & 7;
        const int n = p & 7;
        const int row = k * 8 + m;
        const int rt  = row >> 4;
        const int r16 = row & 15;
        const float c1n  = s_stage[rt][r16][n];        // sum w*xr_m*xr_n
        const float c1n8 = s_stage[rt][r16][n + 8];    // sum w*xr_m*xi_n
        const float c2n  = s_stage[4 + rt][r16][n];    // sum w*xi_m*xr_n
        const float c2n8 = s_stage[4 + rt][r16][n + 8];// sum w*xi_m*xi_n
        s_Vre[k][m][n] = (c1n + c2n8) * (1.0f / T_);
        s_Vim[k][m][n] = (c2n - c1n8) * (1.0f / T_);
    }
    __syncthreads();

    // ---------------- Phase C: diagonal loading ----------------
    if (tid < M_) {
        float tr = 0.0f;
#pragma unroll
        for (int m = 0; m < M_; ++m) tr += s_Vre[tid][m][m];
        const float add = fmaxf(tr, 1.0f) * EPS_;
#pragma unroll
        for (int m = 0; m < M_; ++m) s_Vre[tid][m][m] += add;
    }
    __syncthreads();

    // ---------------- Phase D: ISS sweeps (wave 0 only) ----------------
    if (wave == 0) {
        for (int it = 0; it < N_ITER_; ++it) {
            for (int kp = 0; kp < M_; ++kp) {
                // q = row kp of Q (saved before update)
                float qre[M_], qim[M_];
#pragma unroll
                for (int n = 0; n < M_; ++n) {
                    qre[n] = s_Qr[kp][n];
                    qim[n] = s_Qi[kp][n];
                }
                // Vq[kz][mz] = sum_n V[kz][mz][n] * conj(q[n]); 2 (kz,mz) per lane
#pragma unroll
                for (int h = 0; h < 2; ++h) {
                    const int p  = lane + h * 32;
                    const int kz = p >> 3;
                    const int mz = p & 7;
                    float vr = 0.0f, vi = 0.0f;
#pragma unroll
                    for (int n = 0; n < M_; ++n) {
                        const float ar = s_Vre[kz][mz][n];
                        const float ai = s_Vim[kz][mz][n];
                        vr += ar * qre[n] + ai * qim[n];
                        vi += ai * qre[n] - ar * qim[n];
                    }
                    s_Vqr[kz][mz] = vr;
                    s_Vqi[kz][mz] = vi;
                }
                // lanes 0..7 reduce over m, compute v[kz]
                if (lane < M_) {
                    float qvq = 0.0f, numr = 0.0f, numi = 0.0f;
#pragma unroll
                    for (int m = 0; m < M_; ++m) {
                        const float wr = s_Vqr[lane][m];
                        const float wi = s_Vqi[lane][m];
                        qvq  += qre[m] * wr - qim[m] * wi;
                        const float Qre = s_Qr[lane][m];
                        const float Qim = s_Qi[lane][m];
                        numr += Qre * wr - Qim * wi;
                        numi += Qre * wi + Qim * wr;
                    }
                    qvq = fmaxf(qvq, EPS2_);
                    float vre, vim;
                    if (lane == kp) {
                        vre = 1.0f - rsqrtf(qvq);
                        vim = 0.0f;
                    } else {
                        vre = numr / qvq;
                        vim = numi / qvq;
                    }
                    s_vr[lane] = vre;
                    s_vi[lane] = vim;
                }
                // Q -= v (outer) q
#pragma unroll
                for (int h = 0; h < 2; ++h) {
                    const int p  = lane + h * 32;
                    const int kz = p >> 3;
                    const int mz = p & 7;
                    const float vre = s_vr[kz];
                    const float vim = s_vi[kz];
                    s_Qr[kz][mz] -= vre * qre[mz] - vim * qim[mz];
                    s_Qi[kz][mz] -= vre * qim[mz] + vim * qre[mz];
                }
            }
        }
    }
    __syncthreads();

    // ---------------- Phase E: Qx, xt (float4 stores), partial sums --------
    float lsum = 0.0f;
    {
        v4f* outXT4 = (v4f*)outXT;
        for (int p4 = tid; p4 < MT4_; p4 += 256) {
            const int m  = p4 >> 7;             // 128 float4 per row (T_=512)
            const int tq = (p4 & 127) * 4;
            v4f out;
#pragma unroll
            for (int j = 0; j < 4; ++j) {
                const int t = tq + j;
                float ar = 0.0f, ai = 0.0f;
#pragma unroll
                for (int n = 0; n < M_; ++n) {
                    const float qr = s_Qr[m][n];
                    const float qi = s_Qi[m][n];
                    const float xr = s_xr[n][t];
                    const float xi = s_xi[n][t];
                    ar += qr * xr - qi * xi;
                    ai += qr * xi + qi * xr;
                }
                const float val = ar * ar + ai * ai;
                ((float*)&out)[j] = val;
                lsum += val;
            }
            // regular-temporal store: keep resident in L2 for the norm pass
            outXT4[(size_t)bf * MT4_ + p4] = out;
        }
    }
    s_red[tid] = lsum;
    __syncthreads();
    for (int s = 128; s > 0; s >>= 1) {
        if (tid < s) s_red[tid] += s_red[tid + s];
        __syncthreads();
    }
    if (tid == 0) partial[bf] = s_red[0];

    if (tid < 64) {
        const size_t qo = ((size_t)bf * 64 + tid) * 2;
        outQ[qo]     = s_Qr[tid >> 3][tid & 7];
        outQ[qo + 1] = s_Qi[tid >> 3][tid & 7];
    }
}

// ---------------------------------------------------------------------------
// Kernel 2: deterministic fixed-order reduction of per-(b,f) partials -> scale[b]
// ---------------------------------------------------------------------------
__global__ __launch_bounds__(256) void iss_scale_kernel(
    const float* __restrict__ partial, float* __restrict__ scale)
{
    __shared__ float red[256];
    const int b = blockIdx.x;
    const int tid = threadIdx.x;
    float s = 0.0f;
    for (int f = tid; f < F_; f += 256) s += partial[b * F_ + f];
    red[tid] = s;
    __syncthreads();
    for (int st = 128; st > 0; st >>= 1) {
        if (tid < st) red[tid] += red[tid + st];
        __syncthreads();
    }
    if (tid == 0) scale[b] = red[0] / (float)((size_t)F_ * M_ * T_);
}

// ---------------------------------------------------------------------------
// Kernel 3: normalize xt by scale[b] and Q by rsqrt(max(scale[b],1e-6)).
// Reads hit L2 (written by kernel1); final stores are non-temporal (stream out).
// ---------------------------------------------------------------------------
__global__ __launch_bounds__(256) void iss_norm_kernel(
    float* __restrict__ outQ, float* __restrict__ outXT,
    const float* __restrict__ scale)
{
    const size_t idx = (size_t)blockIdx.x * blockDim.x + threadIdx.x;

    const size_t XT4  = XTN_ / 4;                      // 4,202,496
    const size_t XT4b = (size_t)F_ * M_ * T_ / 4;      // per-batch chunk
    if (idx < XT4) {
        const int b = (int)(idx / XT4b);
        const float inv = 1.0f / scale[b];
        v4f* p = (v4f*)outXT;
        v4f v = p[idx];
        v.x *= inv; v.y *= inv; v.z *= inv; v.w *= inv;
        __builtin_nontemporal_store(v, &p[idx]);
    }

    const size_t Q4  = QN_ / 4;                        // 131,328
    const size_t Q4b = (size_t)F_ * M_ * M_ * 2 / 4;   // per-batch chunk
    if (idx < Q4) {
        const int b = (int)(idx / Q4b);
        const float sc = rsqrtf(fmaxf(scale[b], 1e-6f));
        v4f* p = (v4f*)outQ;
        v4f v = p[idx];
        v.x *= sc; v.y *= sc; v.z *= sc; v.w *= sc;
        __builtin_nontemporal_store(v, &p[idx]);
    }
}

// ---------------------------------------------------------------------------
extern "C" void kernel_launch(void* const* d_in, const int* in_sizes, int n_in,
                              void* d_out, int out_size, void* d_ws, size_t ws_size,
                              hipStream_t stream) {
    (void)in_sizes; (void)n_in; (void)out_size; (void)ws_size;

    const float* g_r  = (const float*)d_in[0];
    const float* g_xr = (const float*)d_in[1];
    const float* g_xi = (const float*)d_in[2];
    const float* g_Qr = (const float*)d_in[3];
    const float* g_Qi = (const float*)d_in[4];

    float* outQ  = (float*)d_out;            // QN_ floats (interleaved complex)
    float* outXT = outQ + QN_;               // XTN_ floats

    float* partial = (float*)d_ws;           // B*F floats
    float* scale   = partial + (B_ * F_);    // B floats

    iss_main_kernel<<<B_ * F_, 256, 0, stream>>>(g_r, g_xr, g_xi, g_Qr, g_Qi,
                                                 outQ, outXT, partial);
    iss_scale_kernel<<<B_, 256, 0, stream>>>(partial, scale);
    const int nblk = (int)((XTN_ / 4 + 255) / 256);
    iss_norm_kernel<<<nblk, 256, 0, stream>>>(outQ, outXT, scale);
}